// PositionalEncoding_8899172237613
// MI455X (gfx1250) — compile-verified
//
#include <hip/hip_runtime.h>
#include <math.h>

// out[b,s,d] = x[b,s,d] * sqrt(512) + PE where
//   PE[..., 2j]   = sin(pos * exp(-ln(1e4) * 2j / 512))
//   PE[..., 2j+1] = cos(pos * exp(-ln(1e4) * 2j / 512))
// Streaming, HBM-bound: 128 MiB traffic -> ~5.6 us floor at 23.3 TB/s.

typedef float v4f __attribute__((ext_vector_type(4)));

#define D_MODEL 512
#define HALF_D  256            // number of (sin,cos) pairs per row
#define ROW_F4  (D_MODEL / 4)  // 128 float4 per row
#define BLOCK   256            // 8 wave32 per workgroup
#define ROWS_PER_ITER 2        // 256 threads / 128 float4-per-row
#define GRID    4096

__global__ __launch_bounds__(BLOCK)
void pe_stream_kernel(const float* __restrict__ x,
                      const int*   __restrict__ pos,
                      float*       __restrict__ out,
                      int n_rows)
{
    // div_term table, built once per block: dt[j] = exp2(j * -log2(10000)/256)
    __shared__ float dt[HALF_D];
    {
        const float K = -0.05190512648261504f;      // -log2(10000) / 256
        dt[threadIdx.x] = __builtin_amdgcn_exp2f((float)threadIdx.x * K); // v_exp_f32
    }
    __syncthreads();

    const float scale = 22.627416997969522f;        // sqrt(512)

    const int t          = threadIdx.x;
    const int row_in_blk = t >> 7;                  // 0..1
    const int col4       = t & (ROW_F4 - 1);        // 0..127
    const int j0         = col4 * 2;                // pair index of .x/.y

    const float dt0 = dt[j0];
    const float dt1 = dt[j0 + 1];

    const int row_stride = (int)gridDim.x * ROWS_PER_ITER;

    for (int base = (int)blockIdx.x * ROWS_PER_ITER; base < n_rows; base += row_stride)
    {
        const int row = base + row_in_blk;
        if (row >= n_rows) continue;

        const v4f* __restrict__ xin = (const v4f*)(x   + (size_t)row * D_MODEL);
        v4f*       __restrict__ op  = (v4f*)      (out + (size_t)row * D_MODEL);

        // Prefetch next tile for this lane (global_prefetch_b8).
        const int nrow = row + row_stride;
        if (nrow < n_rows) {
            __builtin_prefetch(x + (size_t)nrow * D_MODEL + (size_t)col4 * 4, 0, 0);
        }

        // Streamed once: non-temporal b128 load.
        v4f v = __builtin_nontemporal_load(&xin[col4]);

        const float p  = (float)pos[row];           // uniform across 128 lanes
        const float a0 = p * dt0;
        const float a1 = p * dt1;

        float s0, c0, s1, c1;
        __sincosf(a0, &s0, &c0);                    // v_sin_f32 / v_cos_f32 (TRANS)
        __sincosf(a1, &s1, &c1);

        v4f r;
        r.x = fmaf(v.x, scale, s0);
        r.y = fmaf(v.y, scale, c0);
        r.z = fmaf(v.z, scale, s1);
        r.w = fmaf(v.w, scale, c1);

        // Written once: non-temporal b128 store (wr-rinse, spare the 192MB L2).
        __builtin_nontemporal_store(r, &op[col4]);
    }
}

extern "C" void kernel_launch(void* const* d_in, const int* in_sizes, int n_in,
                              void* d_out, int out_size, void* d_ws, size_t ws_size,
                              hipStream_t stream)
{
    const float* x   = (const float*)d_in[0];   // [B, S, 512] fp32
    const int*   pos = (const int*)  d_in[1];   // [B, S] int32
    float*       out = (float*)d_out;           // [B, S, 512] fp32

    const int n_rows = in_sizes[1];             // B*S = 32768

    dim3 grid(GRID);
    dim3 block(BLOCK);
    pe_stream_kernel<<<grid, block, 0, stream>>>(x, pos, out, n_rows);
}